// LuongAttention_8375186227934
// MI455X (gfx1250) — compile-verified
//
#include <hip/hip_runtime.h>
#include <hip/hip_bf16.h>
#include <cstddef>

typedef float v2f __attribute__((ext_vector_type(2)));
typedef float v8f __attribute__((ext_vector_type(8)));

#define B_    4
#define TGT_  256
#define SRC_  256
#define D_    512          // TS == SS == 512
#define WCOLS 1024         // W is (512, 1024)

// ---------------------------------------------------------------------------
// fast tanh:  tanh(x) = 1 - 2/(exp(2x)+1), exp via native v_exp_f32 (exp2)
// ---------------------------------------------------------------------------
__device__ __forceinline__ float tanh_fast(float x) {
    float e = __builtin_amdgcn_exp2f(x * 2.8853900817779268f); // 2*log2(e)
    return 1.0f - 2.0f * __builtin_amdgcn_rcpf(e + 1.0f);
}

// ---------------------------------------------------------------------------
// Kernel 1: projections.  Computes
//   sel==0: hp[m,n] = sum_k hidden[m,k] * W[n, k]          (m = b*T+t)
//   sel==1: ep[m,n] = sum_k enc[m,k]    * W[n, 512+k] + b[n]
// 256 threads = 8 waves; each wave owns one 16x16 output tile via
// V_WMMA_F32_16X16X4_F32, K-loop of 128 steps.
// ---------------------------------------------------------------------------
__global__ __launch_bounds__(256)
void proj_wmma_kernel(const float* __restrict__ hidden,
                      const float* __restrict__ enc,
                      const float* __restrict__ W,
                      const float* __restrict__ bias,
                      float* __restrict__ hp,
                      float* __restrict__ ep) {
    const int id    = blockIdx.x;          // [0, 512)
    const int sel   = id >> 8;             // 0 = hp, 1 = ep
    const int rem   = id & 255;
    const int m0    = (rem >> 2) << 4;     // 64 m-tiles of 16 rows (B*256 rows)
    const int ng    = rem & 3;             // 4 n-groups of 128 cols

    const int lane  = threadIdx.x & 31;
    const int wav   = threadIdx.x >> 5;
    const int n0    = ng * 128 + wav * 16;

    const int half  = lane >> 4;           // 0: lanes 0-15, 1: lanes 16-31
    const int l15   = lane & 15;
    const int kb    = half * 2;            // A/B k-pair owned by this lane

    const float* Arow = (sel ? enc : hidden) + (size_t)(m0 + l15) * D_;
    const float* Wrow = W + (size_t)(n0 + l15) * WCOLS + sel * D_;

    v8f acc = {};
    #pragma unroll 8
    for (int k0 = 0; k0 < D_; k0 += 4) {
        v2f a = *(const v2f*)(Arow + k0 + kb);   // A[m, k0+kb .. +1]
        v2f b = *(const v2f*)(Wrow + k0 + kb);   // B[k0+kb.., n] = W[n, k..]
        acc = __builtin_amdgcn_wmma_f32_16x16x4_f32(
                  false, a, false, b, (short)0, acc, false, false);
    }

    float*       out = sel ? ep : hp;
    const float  bv  = sel ? bias[n0 + l15] : 0.0f;   // fold +b into ep
    #pragma unroll
    for (int j = 0; j < 8; ++j) {
        int row = m0 + j + half * 8;
        out[(size_t)row * D_ + n0 + l15] = acc[j] + bv;
    }
}

// ---------------------------------------------------------------------------
// Kernel 2: score + masked softmax, fused.  One block per (b, t).
//   score[s] = sum_o v[o] * tanh(hp[b,t,o] + ep[b,s,o])    (bias in ep)
// then softmax over s, * mask, renormalize (+1e-12), write probs.
// hp row and v staged in LDS (broadcast reads); ep rows stream from L2.
// ---------------------------------------------------------------------------
__global__ __launch_bounds__(256)
void score_softmax_kernel(const float* __restrict__ hp,
                          const float* __restrict__ ep,
                          const int*   __restrict__ masks,
                          const float* __restrict__ v,
                          float* __restrict__ probs) {
    __shared__ float hrow[D_];
    __shared__ float vsh[D_];
    __shared__ float red[256];

    const int tid = threadIdx.x;
    const int bt  = blockIdx.x;            // b*256 + t
    const int b   = bt >> 8;

    const float* hsrc = hp + (size_t)bt * D_;
    hrow[tid]       = hsrc[tid];
    hrow[tid + 256] = hsrc[tid + 256];
    vsh[tid]        = v[tid];
    vsh[tid + 256]  = v[tid + 256];
    __syncthreads();

    const int s = tid;
    const float* erow = ep + ((size_t)b * SRC_ + s) * D_;
    float acc = 0.0f;
    #pragma unroll 4
    for (int o = 0; o < D_; o += 4) {
        float4 e4 = *(const float4*)(erow + o);
        acc = fmaf(vsh[o + 0], tanh_fast(hrow[o + 0] + e4.x), acc);
        acc = fmaf(vsh[o + 1], tanh_fast(hrow[o + 1] + e4.y), acc);
        acc = fmaf(vsh[o + 2], tanh_fast(hrow[o + 2] + e4.z), acc);
        acc = fmaf(vsh[o + 3], tanh_fast(hrow[o + 3] + e4.w), acc);
    }

    // ---- softmax over the 256 s-values held one-per-thread ----
    red[tid] = acc; __syncthreads();
    for (int st = 128; st > 0; st >>= 1) {
        if (tid < st) red[tid] = fmaxf(red[tid], red[tid + st]);
        __syncthreads();
    }
    const float mx = red[0]; __syncthreads();

    const float e = __builtin_amdgcn_exp2f((acc - mx) * 1.4426950408889634f);
    red[tid] = e; __syncthreads();
    for (int st = 128; st > 0; st >>= 1) {
        if (tid < st) red[tid] += red[tid + st];
        __syncthreads();
    }
    const float esum = red[0]; __syncthreads();

    const float pm = (e / esum) * (float)masks[b * SRC_ + s];
    red[tid] = pm; __syncthreads();
    for (int st = 128; st > 0; st >>= 1) {
        if (tid < st) red[tid] += red[tid + st];
        __syncthreads();
    }
    const float msum = red[0];

    probs[(size_t)bt * SRC_ + s] = pm / (msum + 1e-12f);
}

// ---------------------------------------------------------------------------
// Kernel 3: context[b,t,d] = sum_s probs[b,t,s] * enc[b,s,d]
// Per batch: M=256 (t), N=512 (d), K=256 (s).  f32 WMMA, 8 waves/block,
// one 16x16 tile per wave, K-loop of 64 steps.
// ---------------------------------------------------------------------------
__global__ __launch_bounds__(256)
void context_wmma_kernel(const float* __restrict__ probs,
                         const float* __restrict__ enc,
                         float* __restrict__ ctx) {
    const int id  = blockIdx.x;            // [0, 256)
    const int ng  = id & 3;
    const int mt  = (id >> 2) & 15;
    const int b   = id >> 6;

    const int lane = threadIdx.x & 31;
    const int wav  = threadIdx.x >> 5;
    const int half = lane >> 4;
    const int l15  = lane & 15;
    const int kb   = half * 2;

    const int m0 = mt * 16;
    const int n0 = ng * 128 + wav * 16;

    const float* Arow = probs + ((size_t)b * TGT_ + m0 + l15) * SRC_;
    const float* Bmat = enc + (size_t)b * SRC_ * D_;

    v8f acc = {};
    #pragma unroll 8
    for (int k0 = 0; k0 < SRC_; k0 += 4) {
        v2f a = *(const v2f*)(Arow + k0 + kb);
        v2f bm;
        bm.x = Bmat[(size_t)(k0 + kb + 0) * D_ + n0 + l15];
        bm.y = Bmat[(size_t)(k0 + kb + 1) * D_ + n0 + l15];
        acc = __builtin_amdgcn_wmma_f32_16x16x4_f32(
                  false, a, false, bm, (short)0, acc, false, false);
    }

    #pragma unroll
    for (int j = 0; j < 8; ++j) {
        int row = m0 + j + half * 8;
        ctx[((size_t)b * TGT_ + row) * D_ + n0 + l15] = acc[j];
    }
}

// ---------------------------------------------------------------------------
// launch
// ---------------------------------------------------------------------------
extern "C" void kernel_launch(void* const* d_in, const int* in_sizes, int n_in,
                              void* d_out, int out_size, void* d_ws, size_t ws_size,
                              hipStream_t stream) {
    const float* hidden = (const float*)d_in[0];   // (4,256,512)
    const float* enc    = (const float*)d_in[1];   // (4,256,512)
    const int*   masks  = (const int*)  d_in[2];   // (4,256)
    const float* W      = (const float*)d_in[3];   // (512,1024)
    const float* bias   = (const float*)d_in[4];   // (512,)
    const float* v      = (const float*)d_in[5];   // (512,)

    float* ctx   = (float*)d_out;                                    // 4*256*512
    float* probs = (float*)d_out + (size_t)B_ * TGT_ * D_;           // 4*256*256

    float* hp = (float*)d_ws;                                        // 2 MB
    float* ep = (float*)d_ws + (size_t)B_ * SRC_ * D_;               // 2 MB

    proj_wmma_kernel<<<512, 256, 0, stream>>>(hidden, enc, W, bias, hp, ep);
    score_softmax_kernel<<<B_ * TGT_, 256, 0, stream>>>(hp, ep, masks, v, probs);
    context_wmma_kernel<<<256, 256, 0, stream>>>(probs, enc, ctx);
}